// INEncoder_52939766890760
// MI455X (gfx1250) — compile-verified
//
#include <hip/hip_runtime.h>

// ---------------------------------------------------------------------------
// Interaction-network encoder, fully fused per batch element.
// grid = B workgroups (one batch element each), block = 256 threads = 8 waves.
// Matrix math on V_WMMA_F32_16X16X4_F32 (fp32-exact vs the fp32 reference).
// d_ws requirement: B*C*E*F*4 bytes ~= 260 MB (inter-layer edge activations).
// ---------------------------------------------------------------------------

#define CDIM 2
#define NNODE 32
#define FDIM 64
#define LLAY 3
#define EE 992          // N*(N-1)
#define XP 66           // padded row stride (floats) for x/agg/W1^T
#define HP 68           // padded row stride for h tiles (16B-aligned rows)
#define W0S 194         // padded row stride for W0^T (Kmax=192 + 2)
#define NTHREADS 256
#define NWAVES 8
#define EPSLN 1e-5f

typedef __attribute__((ext_vector_type(2))) float v2f;
typedef __attribute__((ext_vector_type(8))) float v8f;

// One contiguous K segment of D += A(16xK) * B(KxN=64), one wave, 16-row tile.
// A lane row m = lane&15; a[r] = arow[k0 + r + 2*khalf]  (arow pre-offset so
// that global k indices apply directly; may be negatively offset).
// B from LDS W^T stored as [o][k]: b[r] = wt[(n16+16*nt)*wstride + k0+r+2*khalf]
template <int KB, int KE>
__device__ __forceinline__ void gemm_seg(const float* arow,
                                         const float* wt, const int wstride,
                                         v8f acc[4], int n16, int khalf) {
#pragma unroll
  for (int k0 = KB; k0 < KE; k0 += 4) {
    const int kidx = k0 + 2 * khalf;
    v2f a = *(const v2f*)(arow + kidx);
#pragma unroll
    for (int nt = 0; nt < 4; nt++) {
      v2f bv = *(const v2f*)(wt + (n16 + 16 * nt) * wstride + kidx);
      acc[nt] = __builtin_amdgcn_wmma_f32_16x16x4_f32(
          false, a, false, bv, (short)0, acc[nt], false, false);
    }
  }
}

__device__ __forceinline__ void block_reduce2(float& s1, float& s2,
                                              float* red, int tid) {
#pragma unroll
  for (int off = 16; off > 0; off >>= 1) {
    s1 += __shfl_xor(s1, off, 32);
    s2 += __shfl_xor(s2, off, 32);
  }
  const int wid = tid >> 5;
  if ((tid & 31) == 0) { red[2 * wid] = s1; red[2 * wid + 1] = s2; }
  __syncthreads();
  s1 = 0.f; s2 = 0.f;
#pragma unroll
  for (int w = 0; w < NWAVES; w++) { s1 += red[2 * w]; s2 += red[2 * w + 1]; }
  __syncthreads();  // red[] reusable afterwards
}

__global__ void __launch_bounds__(NTHREADS)
ine_encoder_kernel(const float* __restrict__ nodes,
                   const float* __restrict__ e0_w0, const float* __restrict__ e0_b0,
                   const float* __restrict__ e0_w1, const float* __restrict__ e0_b1,
                   const float* __restrict__ eL_w0, const float* __restrict__ eL_b0,
                   const float* __restrict__ eL_w1, const float* __restrict__ eL_b1,
                   const float* __restrict__ n_w0,  const float* __restrict__ n_b0,
                   const float* __restrict__ n_w1,  const float* __restrict__ n_b1,
                   const float* __restrict__ node_ln_w, const float* __restrict__ node_ln_b,
                   const float* __restrict__ edge_ln_w, const float* __restrict__ edge_ln_b,
                   const int* __restrict__ srcg, const int* __restrict__ dstg,
                   float* __restrict__ out, float* __restrict__ e_scr) {
  __shared__ float x_lds[CDIM * NNODE * XP];     // node state (padded rows)
  __shared__ float agg_lds[CDIM * NNODE * XP];   // scatter accumulator
  __shared__ float w0t[FDIM * W0S];              // staged W0^T ([o][k])
  __shared__ float w1t[FDIM * XP];               // staged W1^T ([o][k])
  __shared__ float h_lds[NWAVES * 16 * HP];      // per-wave A-stage / hidden tile
  __shared__ float red[2 * NWAVES];
  __shared__ int src_s[EE];
  __shared__ int dst_s[EE];

  const int tid   = threadIdx.x;
  const int lane  = tid & 31;
  const int wid   = tid >> 5;
  const int n16   = lane & 15;
  const int khalf = lane >> 4;
  const int b     = blockIdx.x;
  const int hb    = wid * 16 * HP;               // this wave's h_lds base
  const int hrow  = hb + n16 * HP;               // this lane's A row in h_lds

  // ---- init: broadcast nodes into both channels of x, stage edge indices
  for (int idx = tid; idx < NNODE * FDIM; idx += NTHREADS) {
    const int n = idx >> 6, f = idx & 63;
    const float v = nodes[(size_t)b * NNODE * FDIM + idx];
    x_lds[n * XP + f] = v;
    x_lds[(NNODE + n) * XP + f] = v;
  }
  for (int idx = tid; idx < EE; idx += NTHREADS) {
    src_s[idx] = srcg[idx];
    dst_s[idx] = dstg[idx];
  }

  float e_mu = 0.f, e_inv = 0.f;   // edge-LN stats carried between layers

  for (int layer = 0; layer < LLAY; layer++) {
    // ---- zero agg
    for (int idx = tid; idx < CDIM * NNODE * XP; idx += NTHREADS) agg_lds[idx] = 0.f;
    __syncthreads();

    const int K1 = (layer == 0) ? 128 : 192;
    float s1 = 0.f, s2 = 0.f;      // edge-LN statistics (this thread)

    // ================= edge MLP, per channel =================
    for (int c = 0; c < CDIM; c++) {
      const float* w0g = (layer == 0) ? (e0_w0 + (size_t)c * FDIM * 128)
                                      : (eL_w0 + (size_t)((layer - 1) * CDIM + c) * FDIM * 192);
      const float* w1g = (layer == 0) ? (e0_w1 + (size_t)c * FDIM * FDIM)
                                      : (eL_w1 + (size_t)((layer - 1) * CDIM + c) * FDIM * FDIM);
      const float* b0g = (layer == 0) ? (e0_b0 + c * FDIM)
                                      : (eL_b0 + ((layer - 1) * CDIM + c) * FDIM);
      const float* b1g = (layer == 0) ? (e0_b1 + c * FDIM)
                                      : (eL_b1 + ((layer - 1) * CDIM + c) * FDIM);

      // stage W0^T / W1^T into padded LDS
      for (int idx = tid; idx < FDIM * K1; idx += NTHREADS) {
        const int o = idx / K1, k = idx - o * K1;
        w0t[o * W0S + k] = w0g[idx];
      }
      for (int idx = tid; idx < FDIM * FDIM; idx += NTHREADS) {
        const int o = idx >> 6, k = idx & 63;
        w1t[o * XP + k] = w1g[idx];
      }
      __syncthreads();

      float b0r[4], b1r[4];
#pragma unroll
      for (int nt = 0; nt < 4; nt++) {
        b0r[nt] = b0g[n16 + 16 * nt];
        b1r[nt] = b1g[n16 + 16 * nt];
      }

      const float* lnw_base = (layer >= 1)
          ? (edge_ln_w + (size_t)((layer - 1) * CDIM + c) * EE * FDIM) : nullptr;
      const float* lnb_base = (layer >= 1)
          ? (edge_ln_b + (size_t)((layer - 1) * CDIM + c) * EE * FDIM) : nullptr;
      float* escr_c = e_scr + ((size_t)b * CDIM + c) * EE * FDIM;

      for (int t = wid; t < EE / 16; t += NWAVES) {
        const int ebase = t * 16;
        const int erow  = ebase + n16;            // this lane's A-matrix row
        const int dn = dst_s[erow], sn = src_s[erow];
        const int xdo = (c * NNODE + dn) * XP;
        const int xso = (c * NNODE + sn) * XP;

        // ---- stage LN(e_prev) A-tile into this wave's h slice (layers >= 1)
        // lane covers row erow, features [khalf*32, khalf*32+32)
        if (layer >= 1) {
          const int fb = khalf * 32;
          const float* ep = escr_c + (size_t)erow * FDIM + fb;
          const float* wp = lnw_base + (size_t)erow * FDIM + fb;
          const float* bp = lnb_base + (size_t)erow * FDIM + fb;
          float* hs = &h_lds[hrow + fb];
#pragma unroll
          for (int j = 0; j < 8; j++) {
            float4 ev = *(const float4*)(ep + 4 * j);
            float4 wv = *(const float4*)(wp + 4 * j);
            float4 bv = *(const float4*)(bp + 4 * j);
            float4 rv;
            rv.x = (ev.x - e_mu) * e_inv * wv.x + bv.x;
            rv.y = (ev.y - e_mu) * e_inv * wv.y + bv.y;
            rv.z = (ev.z - e_mu) * e_inv * wv.z + bv.z;
            rv.w = (ev.w - e_mu) * e_inv * wv.w + bv.w;
            *(float4*)(hs + 4 * j) = rv;
          }
        }

        // ---- GEMM1: concat(x[dst], x[src], LN(e_prev)) @ W0^T (segmented)
        v8f acc[4] = {};
        gemm_seg<0, 64>(&x_lds[xdo], w0t, W0S, acc, n16, khalf);
        gemm_seg<64, 128>(&x_lds[xso] - 64, w0t, W0S, acc, n16, khalf);
        if (layer >= 1)
          gemm_seg<128, 192>(&h_lds[hrow] - 128, w0t, W0S, acc, n16, khalf);

        // ---- h = relu(acc + b0) -> this wave's LDS tile
#pragma unroll
        for (int r = 0; r < 8; r++) {
          const int m = r + 8 * khalf;
#pragma unroll
          for (int nt = 0; nt < 4; nt++) {
            float v = acc[nt][r] + b0r[nt];
            v = v > 0.f ? v : 0.f;
            h_lds[hb + m * HP + n16 + 16 * nt] = v;
          }
        }

        // ---- GEMM2: h @ W1^T
        v8f acc2[4] = {};
        gemm_seg<0, 64>(&h_lds[hrow], w1t, XP, acc2, n16, khalf);

        // ---- epilogue: e = relu(acc2 + b1); scatter-add, LN stats, store
#pragma unroll
        for (int r = 0; r < 8; r++) {
          const int m   = r + 8 * khalf;
          const int ee2 = ebase + m;
          const int agb = (c * NNODE + dst_s[ee2]) * XP;
#pragma unroll
          for (int nt = 0; nt < 4; nt++) {
            float v = acc2[nt][r] + b1r[nt];
            v = v > 0.f ? v : 0.f;
            const int col = n16 + 16 * nt;
            atomicAdd(&agg_lds[agb + col], v);
            if (layer < LLAY - 1) {
              s1 += v; s2 += v * v;
              escr_c[(size_t)ee2 * FDIM + col] = v;
            }
          }
        }
      }
      __threadfence_block();
      __syncthreads();
    }

    // ---- edge-LN statistics for next layer (per batch element)
    if (layer < LLAY - 1) {
      block_reduce2(s1, s2, red, tid);
      const float cnt = (float)(CDIM * EE * FDIM);
      e_mu = s1 / cnt;
      const float var = s2 / cnt - e_mu * e_mu;
      e_inv = rsqrtf(var + EPSLN);
    }

    // ================= node MLP, per channel =================
    for (int c = 0; c < CDIM; c++) {
      const float* w0g = n_w0 + (size_t)(layer * CDIM + c) * FDIM * 128;
      const float* w1g = n_w1 + (size_t)(layer * CDIM + c) * FDIM * FDIM;
      const float* b0g = n_b0 + (layer * CDIM + c) * FDIM;
      const float* b1g = n_b1 + (layer * CDIM + c) * FDIM;
      for (int idx = tid; idx < FDIM * 128; idx += NTHREADS) {
        const int o = idx >> 7, k = idx & 127;
        w0t[o * W0S + k] = w0g[idx];
      }
      for (int idx = tid; idx < FDIM * FDIM; idx += NTHREADS) {
        const int o = idx >> 6, k = idx & 63;
        w1t[o * XP + k] = w1g[idx];
      }
      __syncthreads();

      if (wid < 2) {                     // 2 waves cover the 32 node rows
        float b0r[4], b1r[4];
#pragma unroll
        for (int nt = 0; nt < 4; nt++) {
          b0r[nt] = b0g[n16 + 16 * nt];
          b1r[nt] = b1g[n16 + 16 * nt];
        }
        const int mt   = wid;
        const int nrow = mt * 16 + n16;
        const int xro  = (c * NNODE + nrow) * XP;

        v8f acc[4] = {};
        gemm_seg<0, 64>(&x_lds[xro], w0t, W0S, acc, n16, khalf);
        gemm_seg<64, 128>(&agg_lds[xro] - 64, w0t, W0S, acc, n16, khalf);

#pragma unroll
        for (int r = 0; r < 8; r++) {
          const int m = r + 8 * khalf;
#pragma unroll
          for (int nt = 0; nt < 4; nt++) {
            float v = acc[nt][r] + b0r[nt];
            v = v > 0.f ? v : 0.f;
            h_lds[hb + m * HP + n16 + 16 * nt] = v;
          }
        }

        v8f acc2[4] = {};
        gemm_seg<0, 64>(&h_lds[hrow], w1t, XP, acc2, n16, khalf);

        // write new x rows (safe: this wave was sole reader of these rows)
#pragma unroll
        for (int r = 0; r < 8; r++) {
          const int nout = mt * 16 + r + 8 * khalf;
#pragma unroll
          for (int nt = 0; nt < 4; nt++) {
            float v = acc2[nt][r] + b1r[nt];
            v = v > 0.f ? v : 0.f;
            x_lds[(c * NNODE + nout) * XP + n16 + 16 * nt] = v;
          }
        }
      }
      __syncthreads();
    }

    // ================= node LayerNorm (over C,N,F per batch element) ======
    {
      float t1 = 0.f, t2 = 0.f;
      for (int idx = tid; idx < CDIM * NNODE * FDIM; idx += NTHREADS) {
        const int cn = idx >> 6, f = idx & 63;
        const float v = x_lds[cn * XP + f];
        t1 += v; t2 += v * v;
      }
      block_reduce2(t1, t2, red, tid);
      const float cnt = (float)(CDIM * NNODE * FDIM);
      const float mu  = t1 / cnt;
      const float inv = rsqrtf(t2 / cnt - mu * mu + EPSLN);
      const size_t lnoff = (size_t)layer * CDIM * NNODE * FDIM;
      for (int idx = tid; idx < CDIM * NNODE * FDIM; idx += NTHREADS) {
        const int cn = idx >> 6, f = idx & 63;
        const float v = x_lds[cn * XP + f];
        x_lds[cn * XP + f] = (v - mu) * inv * node_ln_w[lnoff + idx] + node_ln_b[lnoff + idx];
      }
      __syncthreads();
    }
  }

  // ---- output: sum over channels
  for (int idx = tid; idx < NNODE * FDIM / 2; idx += NTHREADS) {
    const int fi = idx * 2;
    const int n = fi >> 6, f = fi & 63;
    v2f a = *(const v2f*)&x_lds[n * XP + f];
    v2f c = *(const v2f*)&x_lds[(NNODE + n) * XP + f];
    v2f o; o.x = a.x + c.x; o.y = a.y + c.y;
    *(v2f*)&out[(size_t)b * NNODE * FDIM + fi] = o;
  }
}

extern "C" void kernel_launch(void* const* d_in, const int* in_sizes, int n_in,
                              void* d_out, int out_size, void* d_ws, size_t ws_size,
                              hipStream_t stream) {
  (void)n_in; (void)out_size; (void)ws_size;  // ws must be >= B*C*E*F*4 bytes
  const float* nodes     = (const float*)d_in[0];
  const float* e0_w0     = (const float*)d_in[1];
  const float* e0_b0     = (const float*)d_in[2];
  const float* e0_w1     = (const float*)d_in[3];
  const float* e0_b1     = (const float*)d_in[4];
  const float* eL_w0     = (const float*)d_in[5];
  const float* eL_b0     = (const float*)d_in[6];
  const float* eL_w1     = (const float*)d_in[7];
  const float* eL_b1     = (const float*)d_in[8];
  const float* n_w0      = (const float*)d_in[9];
  const float* n_b0      = (const float*)d_in[10];
  const float* n_w1      = (const float*)d_in[11];
  const float* n_b1      = (const float*)d_in[12];
  const float* node_ln_w = (const float*)d_in[13];
  const float* node_ln_b = (const float*)d_in[14];
  const float* edge_ln_w = (const float*)d_in[15];
  const float* edge_ln_b = (const float*)d_in[16];
  const int*   srcg      = (const int*)d_in[17];
  const int*   dstg      = (const int*)d_in[18];

  const int B = in_sizes[0] / (NNODE * FDIM);   // 512
  hipLaunchKernelGGL(ine_encoder_kernel, dim3(B), dim3(NTHREADS), 0, stream,
                     nodes, e0_w0, e0_b0, e0_w1, e0_b1,
                     eL_w0, eL_b0, eL_w1, eL_b1,
                     n_w0, n_b0, n_w1, n_b1,
                     node_ln_w, node_ln_b, edge_ln_w, edge_ln_b,
                     srcg, dstg, (float*)d_out, (float*)d_ws);
}